// GCN_C_PyG_24721831756229
// MI455X (gfx1250) — compile-verified
//
#include <hip/hip_runtime.h>
#include <hip/hip_bf16.h>
#include <math.h>

typedef __attribute__((ext_vector_type(2))) float v2f;
typedef __attribute__((ext_vector_type(8))) float v8f;

#define IN_CH 256
#define HID   16
#define OUTC  40
#define OUTP  48   // padded to 3 tiles of 16

// ---------------------------------------------------------------------------
// 1) degree accumulation: deg[col[e]] += w[e]
// ---------------------------------------------------------------------------
__global__ __launch_bounds__(256) void gcn_degree(const int* __restrict__ cols,
                                                  const float* __restrict__ w,
                                                  float* __restrict__ deg, int nE) {
    int e = blockIdx.x * 256 + threadIdx.x;
    if (e < nE) atomicAdd(&deg[cols[e]], w[e]);
}

// dinv[n] = rsqrt(deg[n] + 1)   (self-loop weight 1; deg >= 1 always)
__global__ __launch_bounds__(256) void gcn_dinv(float* __restrict__ deg, int n) {
    int i = blockIdx.x * 256 + threadIdx.x;
    if (i < n) deg[i] = rsqrtf(deg[i] + 1.0f);
}

// ---------------------------------------------------------------------------
// 2) GEMM1: h1[N,16] = x[N,256] @ W1[256,16]  via V_WMMA_F32_16X16X4_F32
//    One wave per 16-row M tile; N=16 is exactly one tile; K looped by 4.
// ---------------------------------------------------------------------------
__global__ __launch_bounds__(256) void gcn_gemm1_wmma(const float* __restrict__ x,
                                                      const float* __restrict__ W1,
                                                      float* __restrict__ h1, int nNodes) {
    __shared__ float w1s[IN_CH * HID];          // 16 KB
    int tid = threadIdx.x;
    for (int i = tid; i < IN_CH * HID; i += 256) w1s[i] = W1[i];
    __syncthreads();

    int wave  = tid >> 5;                       // 8 waves per block
    int lane  = tid & 31;
    int tileM = blockIdx.x * 8 + wave;
    if (tileM * 16 >= nNodes) return;           // wave-uniform -> EXEC stays all-1s

    int m    = lane & 15;                       // M index for A frag / N index for B frag
    int koff = (lane >> 4) << 1;                // lanes 0-15: K+{0,1}; lanes 16-31: K+{2,3}
    int gm   = tileM * 16 + m;
    if (gm >= nNodes) gm = nNodes - 1;          // clamp (stores guarded below)
    const float* xrow = x + (size_t)gm * IN_CH;

    v8f acc = {};
    #pragma unroll 4
    for (int k = 0; k < IN_CH; k += 4) {
        __builtin_prefetch(xrow + k + 32, 0, 3);      // global_prefetch_b8, near scope
        v2f a, b;
        a.x = xrow[k + koff];
        a.y = xrow[k + koff + 1];
        b.x = w1s[(k + koff) * HID + m];
        b.y = w1s[(k + koff + 1) * HID + m];
        acc = __builtin_amdgcn_wmma_f32_16x16x4_f32(false, a, false, b,
                                                    (short)0, acc, false, false);
    }
    // C/D layout: VGPR r -> M = r + (lane>=16 ? 8 : 0), N = lane&15
    int nn      = lane & 15;
    int moff    = (lane >> 4) * 8;
    int rowBase = tileM * 16;
    float* p = h1 + (size_t)(rowBase + moff) * HID + nn;
    if (rowBase + 16 <= nNodes) {               // wave-uniform full-tile fast path
        #pragma unroll
        for (int r = 0; r < 8; ++r) p[r * HID] = acc[r];
    } else {
        #pragma unroll
        for (int r = 0; r < 8; ++r)
            if (rowBase + moff + r < nNodes) p[r * HID] = acc[r];
    }
}

// ---------------------------------------------------------------------------
// 3) edge scatter, layer 1: agg1[col, j] += dinv[row]*w*dinv[col] * h1[row, j]
//    16 lanes per edge (j = channel)
// ---------------------------------------------------------------------------
__global__ __launch_bounds__(256) void gcn_edge16(const int* __restrict__ rows,
                                                  const int* __restrict__ cols,
                                                  const float* __restrict__ w,
                                                  const float* __restrict__ dinv,
                                                  const float* __restrict__ h,
                                                  float* __restrict__ agg, int nE) {
    int tid = threadIdx.x;
    int e   = blockIdx.x * 16 + (tid >> 4);
    if (e >= nE) return;
    int j = tid & 15;
    int r = rows[e], c = cols[e];
    float nrm = dinv[r] * w[e] * dinv[c];
    atomicAdd(&agg[(size_t)c * HID + j], nrm * h[(size_t)r * HID + j]);
}

// fused: self-loop term + bias + ReLU (in place on agg1)
__global__ __launch_bounds__(256) void gcn_relu_self16(const float* __restrict__ h1,
                                                       const float* __restrict__ dinv,
                                                       const float* __restrict__ b1,
                                                       float* __restrict__ agg, int n) {
    int t = blockIdx.x * 256 + threadIdx.x;
    if (t >= n * HID) return;
    int node = t >> 4, j = t & 15;
    float di = dinv[node];
    float v  = agg[t] + di * di * h1[t] + b1[j];
    agg[t] = v > 0.0f ? v : 0.0f;
}

// ---------------------------------------------------------------------------
// 4) GEMM2: h2[N,40] = relu1[N,16] @ W2[16,40]  (W2 zero-padded to 16x48 in LDS)
// ---------------------------------------------------------------------------
__global__ __launch_bounds__(256) void gcn_gemm2_wmma(const float* __restrict__ hin,
                                                      const float* __restrict__ W2,
                                                      float* __restrict__ h2, int nNodes) {
    __shared__ float w2s[HID * OUTP];           // 3 KB
    int tid = threadIdx.x;
    for (int i = tid; i < HID * OUTP; i += 256) {
        int kk = i / OUTP, n = i - kk * OUTP;
        w2s[i] = (n < OUTC) ? W2[kk * OUTC + n] : 0.0f;
    }
    __syncthreads();

    int wave  = tid >> 5;
    int lane  = tid & 31;
    int tileM = blockIdx.x * 8 + wave;
    if (tileM * 16 >= nNodes) return;           // wave-uniform

    int m    = lane & 15;
    int koff = (lane >> 4) << 1;
    int gm   = tileM * 16 + m;
    if (gm >= nNodes) gm = nNodes - 1;
    const float* arow = hin + (size_t)gm * HID;

    v2f afrag[4];
    #pragma unroll
    for (int ks = 0; ks < 4; ++ks) {
        afrag[ks].x = arow[ks * 4 + koff];
        afrag[ks].y = arow[ks * 4 + koff + 1];
    }

    int nn      = lane & 15;
    int moff    = (lane >> 4) * 8;
    int rowBase = tileM * 16;
    bool full   = (rowBase + 16 <= nNodes);     // wave-uniform

    #pragma unroll
    for (int nt = 0; nt < 3; ++nt) {
        v8f acc = {};
        #pragma unroll
        for (int ks = 0; ks < 4; ++ks) {
            v2f b;
            b.x = w2s[(ks * 4 + koff) * OUTP + nt * 16 + nn];
            b.y = w2s[(ks * 4 + koff + 1) * OUTP + nt * 16 + nn];
            acc = __builtin_amdgcn_wmma_f32_16x16x4_f32(false, afrag[ks], false, b,
                                                        (short)0, acc, false, false);
        }
        int gn = nt * 16 + nn;
        float* p = h2 + (size_t)(rowBase + moff) * OUTC + gn;
        if (full) {
            if (gn < OUTC) {                    // single cmpx, 8 immediate-offset stores
                #pragma unroll
                for (int r = 0; r < 8; ++r) p[r * OUTC] = acc[r];
            }
        } else {
            #pragma unroll
            for (int r = 0; r < 8; ++r)
                if (gn < OUTC && rowBase + moff + r < nNodes) p[r * OUTC] = acc[r];
        }
    }
}

// ---------------------------------------------------------------------------
// 5) edge scatter, layer 2: 40 lanes per edge; 320-thread blocks (8 edges each)
// ---------------------------------------------------------------------------
__global__ __launch_bounds__(320) void gcn_edge40(const int* __restrict__ rows,
                                                  const int* __restrict__ cols,
                                                  const float* __restrict__ w,
                                                  const float* __restrict__ dinv,
                                                  const float* __restrict__ h,
                                                  float* __restrict__ agg, int nE) {
    int tid = threadIdx.x;
    int le  = tid / OUTC;                       // constant division -> mul/shift
    int j   = tid - le * OUTC;
    int e   = blockIdx.x * 8 + le;
    if (e >= nE) return;
    int r = rows[e], c = cols[e];
    float nrm = dinv[r] * w[e] * dinv[c];
    atomicAdd(&agg[(size_t)c * OUTC + j], nrm * h[(size_t)r * OUTC + j]);
}

// fused: self-loop + bias + log_softmax -> d_out
__global__ __launch_bounds__(128) void gcn_logsoftmax(const float* __restrict__ agg2,
                                                      const float* __restrict__ h2,
                                                      const float* __restrict__ dinv,
                                                      const float* __restrict__ b2,
                                                      float* __restrict__ out, int n) {
    int node = blockIdx.x * 128 + threadIdx.x;
    if (node >= n) return;
    float di = dinv[node], d2 = di * di;
    float v[OUTC];
    float mx = -3.4e38f;
    size_t base = (size_t)node * OUTC;
    #pragma unroll
    for (int j = 0; j < OUTC; ++j) {
        float t = agg2[base + j] + d2 * h2[base + j] + b2[j];
        v[j] = t;
        mx = fmaxf(mx, t);
    }
    float s = 0.0f;
    #pragma unroll
    for (int j = 0; j < OUTC; ++j) s += expf(v[j] - mx);
    float lse = logf(s);
    #pragma unroll
    for (int j = 0; j < OUTC; ++j) out[base + j] = v[j] - mx - lse;
}

// ---------------------------------------------------------------------------
extern "C" void kernel_launch(void* const* d_in, const int* in_sizes, int n_in,
                              void* d_out, int out_size, void* d_ws, size_t ws_size,
                              hipStream_t stream) {
    const float* x    = (const float*)d_in[0];
    const int*   ei   = (const int*)d_in[1];     // [2, E] flat
    const float* ew   = (const float*)d_in[2];
    const float* W1   = (const float*)d_in[3];
    const float* b1   = (const float*)d_in[4];
    const float* W2   = (const float*)d_in[5];
    const float* b2   = (const float*)d_in[6];
    float*       out  = (float*)d_out;

    const int nNodes = in_sizes[0] / IN_CH;      // 100000
    const int nE     = in_sizes[2];              // 3200000
    const int* rows  = ei;
    const int* cols  = ei + nE;

    // workspace layout (floats)
    float* dinv = (float*)d_ws;                  // nNodes
    float* h1   = dinv + nNodes;                 // nNodes*16
    float* agg1 = h1   + (size_t)nNodes * HID;   // nNodes*16
    float* h2   = agg1 + (size_t)nNodes * HID;   // nNodes*40
    float* agg2 = h2   + (size_t)nNodes * OUTC;  // nNodes*40

    // zero accumulators each call (graph-capture-safe async memsets)
    hipMemsetAsync(dinv, 0, (size_t)nNodes * sizeof(float), stream);
    hipMemsetAsync(agg1, 0, (size_t)nNodes * HID * sizeof(float), stream);
    hipMemsetAsync(agg2, 0, (size_t)nNodes * OUTC * sizeof(float), stream);

    // 1) degree + dinv
    gcn_degree<<<(nE + 255) / 256, 256, 0, stream>>>(cols, ew, dinv, nE);
    gcn_dinv<<<(nNodes + 255) / 256, 256, 0, stream>>>(dinv, nNodes);

    // 2) layer-1 GEMM (WMMA f32)
    int mTiles = (nNodes + 15) / 16;
    gcn_gemm1_wmma<<<(mTiles + 7) / 8, 256, 0, stream>>>(x, W1, h1, nNodes);

    // 3) layer-1 scatter + fused self-loop/bias/ReLU
    gcn_edge16<<<(nE + 15) / 16, 256, 0, stream>>>(rows, cols, ew, dinv, h1, agg1, nE);
    gcn_relu_self16<<<((size_t)nNodes * HID + 255) / 256, 256, 0, stream>>>(h1, dinv, b1, agg1, nNodes);

    // 4) layer-2 GEMM (WMMA f32)
    gcn_gemm2_wmma<<<(mTiles + 7) / 8, 256, 0, stream>>>(agg1, W2, h2, nNodes);

    // 5) layer-2 scatter + fused self-loop/bias/log_softmax
    gcn_edge40<<<(nE + 7) / 8, 320, 0, stream>>>(rows, cols, ew, dinv, h2, agg2, nE);
    gcn_logsoftmax<<<(nNodes + 127) / 128, 128, 0, stream>>>(agg2, h2, dinv, b2, out, nNodes);
}